// ParallelHyenaOperator_481036337262
// MI455X (gfx1250) — compile-verified
//
#include <hip/hip_runtime.h>
#include <hip/hip_bf16.h>

// ---------------------------------------------------------------------------
// ParallelHyenaOperator on MI455X (gfx1250)
//
// y = causal_conv_L(kv, h_g) + kv*bias ; z = x1 * y ; kv = x2*v
// Conv realized as 16384-pt FFT via Monarch decomposition 16384 = 128*128:
// each FFT stage is a 128x128(x128) fp32 matmul on V_WMMA_F32_16X16X4_F32.
// Complex matmul stages are fused so every operand fragment is loaded once
// and feeds all WMMA ops that consume it (halves LDS/global operand traffic).
// ---------------------------------------------------------------------------

typedef __attribute__((ext_vector_type(2))) float v2f;
typedef __attribute__((ext_vector_type(8))) float v8f;

#define B_  2
#define L_  8192
#define G_  256
#define DG_ 8
#define D_  2048            // G_*DG_
#define N1_ 128             // FFT radix
#define NFFT_ 16384         // 2*L_
#define LP  132             // LDS pitch in floats (bank-conflict avoidance)

// A 16x4 frag: lanes0-15 K={k0,k0+1}, lanes16-31 K={k0+2,k0+3}
// B 4x16 frag: vgpr0 rows k0 / k0+2, vgpr1 rows k0+1 / k0+3
// C/D tile:    vgpr r -> row m0+r (lanes0-15) / m0+8+r (lanes16-31), col=lane&15

#define WMMA_F32(a, b, c) \
  __builtin_amdgcn_wmma_f32_16x16x4_f32(false, (a), false, (b), (short)0, (c), false, false)

template <int NT>
__device__ inline void zero_acc(v8f* acc) {
#pragma unroll
  for (int nt = 0; nt < NT; ++nt)
#pragma unroll
    for (int r = 0; r < 8; ++r) acc[nt][r] = 0.0f;
}

template <int NT>
__device__ inline void store_acc(float* __restrict__ C, int ldc,
                                 const v8f* acc, int m0, int lane) {
  const int half = lane >> 4;
  const int col0 = lane & 15;
#pragma unroll
  for (int nt = 0; nt < NT; ++nt)
#pragma unroll
    for (int r = 0; r < 8; ++r)
      C[(m0 + r + 8 * half) * ldc + nt * 16 + col0] = acc[nt][r];
}

// forward stage 1: real X (LDS) times complex F (global):
//   accR += X*Fc ; accI += X*Fs       (one X fragment drives both)
template <int NT>
__device__ inline void mm_fwd1(v8f* accR, v8f* accI,
                               const float* __restrict__ X, int lda,
                               const float* __restrict__ Fc,
                               const float* __restrict__ Fs, int ldb,
                               int m0, int K, int lane) {
  const int mrow = m0 + (lane & 15);
  const int h2 = (lane >> 4) * 2;
  const int col0 = lane & 15;
  for (int k0 = 0; k0 < K; k0 += 4) {
    v2f a;
    a.x = X[mrow * lda + k0 + h2];
    a.y = X[mrow * lda + k0 + h2 + 1];
#pragma unroll
    for (int nt = 0; nt < NT; ++nt) {
      const int col = nt * 16 + col0;
      v2f bc, bs;
      bc.x = Fc[(k0 + h2) * ldb + col];
      bc.y = Fc[(k0 + h2 + 1) * ldb + col];
      bs.x = Fs[(k0 + h2) * ldb + col];
      bs.y = Fs[(k0 + h2 + 1) * ldb + col];
      accR[nt] = WMMA_F32(a, bc, accR[nt]);
      accI[nt] = WMMA_F32(a, bs, accI[nt]);
    }
  }
}

// complex F (global, symmetric) times complex B (LDS), fused 4-WMMA form:
//   accR += Fc*Br + s*Fs*Bi ; accI += Fc*Bi - s*Fs*Br
//   forward DFT: s = -1 ; conjugate (inverse) DFT: s = +1
template <int NT>
__device__ inline void mm_cplx(v8f* accR, v8f* accI,
                               const float* __restrict__ Fc,
                               const float* __restrict__ Fs, int lda,
                               const float* __restrict__ BR,
                               const float* __restrict__ BI, int ldb,
                               int m0, int K, float s, int lane) {
  const int mrow = m0 + (lane & 15);
  const int h2 = (lane >> 4) * 2;
  const int col0 = lane & 15;
  for (int k0 = 0; k0 < K; k0 += 4) {
    v2f ac, asP, asM;
    ac.x = Fc[mrow * lda + k0 + h2];
    ac.y = Fc[mrow * lda + k0 + h2 + 1];
    float fx = Fs[mrow * lda + k0 + h2];
    float fy = Fs[mrow * lda + k0 + h2 + 1];
    asP.x = s * fx;  asP.y = s * fy;     // +s*Fs
    asM.x = -asP.x;  asM.y = -asP.y;     // -s*Fs
#pragma unroll
    for (int nt = 0; nt < NT; ++nt) {
      const int col = nt * 16 + col0;
      v2f br, bi;
      br.x = BR[(k0 + h2) * ldb + col];
      br.y = BR[(k0 + h2 + 1) * ldb + col];
      bi.x = BI[(k0 + h2) * ldb + col];
      bi.y = BI[(k0 + h2 + 1) * ldb + col];
      accR[nt] = WMMA_F32(ac, br, accR[nt]);
      accR[nt] = WMMA_F32(asP, bi, accR[nt]);
      accI[nt] = WMMA_F32(ac, bi, accI[nt]);
      accI[nt] = WMMA_F32(asM, br, accI[nt]);
    }
  }
}

// inverse stage 2 (real output): acc += Br*Fc + Bi*Fs
template <int NT>
__device__ inline void mm_out(v8f* acc,
                              const float* __restrict__ BR,
                              const float* __restrict__ BI, int lda,
                              const float* __restrict__ Fc,
                              const float* __restrict__ Fs, int ldb,
                              int m0, int K, int lane) {
  const int mrow = m0 + (lane & 15);
  const int h2 = (lane >> 4) * 2;
  const int col0 = lane & 15;
  for (int k0 = 0; k0 < K; k0 += 4) {
    v2f ar, ai;
    ar.x = BR[mrow * lda + k0 + h2];
    ar.y = BR[mrow * lda + k0 + h2 + 1];
    ai.x = BI[mrow * lda + k0 + h2];
    ai.y = BI[mrow * lda + k0 + h2 + 1];
#pragma unroll
    for (int nt = 0; nt < NT; ++nt) {
      const int col = nt * 16 + col0;
      v2f bc, bs;
      bc.x = Fc[(k0 + h2) * ldb + col];
      bc.y = Fc[(k0 + h2 + 1) * ldb + col];
      bs.x = Fs[(k0 + h2) * ldb + col];
      bs.y = Fs[(k0 + h2 + 1) * ldb + col];
      acc[nt] = WMMA_F32(ar, bc, acc[nt]);
      acc[nt] = WMMA_F32(ai, bs, acc[nt]);
    }
  }
}

// ---------------------------------------------------------------------------
// tables: F = exp(-2*pi*i*n*k/128) = (Fc, Fs);  Tw = exp(-2*pi*i*n1*k2/16384)
// ---------------------------------------------------------------------------
__global__ void make_tables(float* __restrict__ Fc, float* __restrict__ Fs,
                            float* __restrict__ Twc, float* __restrict__ Tws) {
  int e = blockIdx.x * 256 + threadIdx.x;
  if (e < N1_ * N1_) {
    int n = e >> 7, k = e & 127;
    float a1 = (float)((n * k) & 127) * (6.283185307179586f / 128.0f);
    Fc[e] = cosf(a1);
    Fs[e] = -sinf(a1);
    float a2 = (float)(n * k) * (6.283185307179586f / 16384.0f);
    Twc[e] = cosf(a2);
    Tws[e] = -sinf(a2);
  }
}

// ---------------------------------------------------------------------------
// kv = x2*v fused with (B,L,D) -> (B,D,L) transpose
// ---------------------------------------------------------------------------
__global__ void kv_transpose(const float* __restrict__ x2,
                             const float* __restrict__ v,
                             float* __restrict__ kv) {
  __shared__ float tile[32][33];
  const int b = blockIdx.z;
  const int d0 = blockIdx.x * 32, l0 = blockIdx.y * 32;
  const size_t base = (size_t)b * L_ * D_;
  for (int r = threadIdx.y; r < 32; r += 8) {
    size_t idx = base + (size_t)(l0 + r) * D_ + d0 + threadIdx.x;
    tile[r][threadIdx.x] = x2[idx] * v[idx];
  }
  __syncthreads();
  for (int r = threadIdx.y; r < 32; r += 8)
    kv[base + (size_t)(d0 + r) * L_ + l0 + threadIdx.x] = tile[threadIdx.x][r];
}

// ---------------------------------------------------------------------------
// one workgroup = one sequence.  mode 0: forward FFT of h row -> (Hfr,Hfi).
// mode 1: forward FFT of kv row, pointwise * Hf[g], inverse FFT, +kv*bias -> Y.
// ---------------------------------------------------------------------------
__global__ __launch_bounds__(256) void fftconv(
    int mode, const float* __restrict__ src,
    const float* __restrict__ Fc, const float* __restrict__ Fs,
    const float* __restrict__ Twc, const float* __restrict__ Tws,
    float* __restrict__ Hfr, float* __restrict__ Hfi,
    const float* __restrict__ bias, float* __restrict__ Yout) {
  __shared__ float smem[4 * N1_ * LP];   // Ar, Ai, Br, Bi complex planes
  float* Ar = smem;
  float* Ai = Ar + N1_ * LP;
  float* Br = Ai + N1_ * LP;
  float* Bi = Br + N1_ * LP;

  const int tid = threadIdx.x;
  const int lane = tid & 31;
  const int w = tid >> 5;          // 8 waves
  const int m0 = w * 16;
  const int seq = blockIdx.x;
  const float* row = src + (size_t)seq * L_;

  // load row as X[n1][n2], n = n1 + 128*n2; columns n2>=64 are the zero pad
  for (int l = tid; l < L_; l += 256)
    Ar[(l & 127) * LP + (l >> 7)] = row[l];
  __syncthreads();

  v8f aR[8], aI[8];

  // forward stage 1 (DFT over n2, K=64 since zero-padded):  B = X * F
  zero_acc<8>(aR);
  zero_acc<8>(aI);
  mm_fwd1<8>(aR, aI, Ar, LP, Fc, Fs, N1_, m0, 64, lane);
  store_acc<8>(Br, LP, aR, m0, lane);
  store_acc<8>(Bi, LP, aI, m0, lane);
  __syncthreads();

  // twiddle: B *= exp(-2pi i n1 k2 / 16384)
  for (int e = tid; e < N1_ * N1_; e += 256) {
    int la = (e >> 7) * LP + (e & 127);
    float tr = Twc[e], ti = Tws[e];
    float br = Br[la], bi = Bi[la];
    Br[la] = br * tr - bi * ti;
    Bi[la] = br * ti + bi * tr;
  }
  __syncthreads();

  // forward stage 2 (DFT over n1, F symmetric):  A = F * B   (fused complex)
  zero_acc<8>(aR);
  zero_acc<8>(aI);
  mm_cplx<8>(aR, aI, Fc, Fs, N1_, Br, Bi, LP, m0, N1_, -1.0f, lane);
  store_acc<8>(Ar, LP, aR, m0, lane);
  store_acc<8>(Ai, LP, aI, m0, lane);
  __syncthreads();

  if (mode == 0) {  // store filter spectrum, indexed e = k1*128 + k2
    float* hr = Hfr + (size_t)seq * (N1_ * N1_);
    float* hi = Hfi + (size_t)seq * (N1_ * N1_);
    for (int e = tid; e < N1_ * N1_; e += 256) {
      int la = (e >> 7) * LP + (e & 127);
      hr[e] = Ar[la];
      hi[e] = Ai[la];
    }
    return;
  }

  // pointwise complex multiply with group's filter spectrum
  const int d = seq & (D_ - 1);
  const int g = d >> 3;  // DG_ = 8
  const float* hr = Hfr + (size_t)g * (N1_ * N1_);
  const float* hi = Hfi + (size_t)g * (N1_ * N1_);
  for (int e = tid; e < N1_ * N1_; e += 256) {
    int la = (e >> 7) * LP + (e & 127);
    float ar = Ar[la], ai = Ai[la], xr = hr[e], xi = hi[e];
    Ar[la] = ar * xr - ai * xi;
    Ai[la] = ar * xi + ai * xr;
  }
  __syncthreads();

  // inverse stage 1:  B = conj(F) * A   (fused complex, s=+1)
  zero_acc<8>(aR);
  zero_acc<8>(aI);
  mm_cplx<8>(aR, aI, Fc, Fs, N1_, Ar, Ai, LP, m0, N1_, 1.0f, lane);
  store_acc<8>(Br, LP, aR, m0, lane);
  store_acc<8>(Bi, LP, aI, m0, lane);
  __syncthreads();

  // conjugate twiddle
  for (int e = tid; e < N1_ * N1_; e += 256) {
    int la = (e >> 7) * LP + (e & 127);
    float tr = Twc[e], ti = Tws[e];
    float br = Br[la], bi = Bi[la];
    Br[la] = br * tr + bi * ti;
    Bi[la] = bi * tr - br * ti;
  }
  __syncthreads();

  // inverse stage 2, real part only, output cols n2 = 0..63 (first L samples):
  //   y = Re( B * conj(F) ) = Br*Fc + Bi*Fs
  v8f acc4[4];
  zero_acc<4>(acc4);
  mm_out<4>(acc4, Br, Bi, LP, Fc, Fs, N1_, m0, N1_, lane);
  store_acc<4>(Ar, LP, acc4, m0, lane);
  __syncthreads();

  // write Y[b][d][l] = y/NFFT + kv*bias   (coalesced: l contiguous in n1)
  const float cb = bias[d];
  float* y = Yout + (size_t)seq * L_;
  for (int l = tid; l < L_; l += 256) {
    int n1 = l & 127, n2 = l >> 7;
    y[l] = Ar[n1 * LP + n2] * (1.0f / 16384.0f) + row[l] * cb;
  }
}

// ---------------------------------------------------------------------------
// z[b,l,d] = x1[b,l,d] * Y[b,d,l]   (transpose-gate)
// ---------------------------------------------------------------------------
__global__ void gate_out(const float* __restrict__ x1,
                         const float* __restrict__ Y,
                         float* __restrict__ z) {
  __shared__ float tile[32][33];
  const int b = blockIdx.z;
  const int d0 = blockIdx.x * 32, l0 = blockIdx.y * 32;
  const size_t base = (size_t)b * L_ * D_;
  for (int r = threadIdx.y; r < 32; r += 8)
    tile[r][threadIdx.x] = Y[base + (size_t)(d0 + r) * L_ + l0 + threadIdx.x];
  __syncthreads();
  for (int r = threadIdx.y; r < 32; r += 8) {
    size_t idx = base + (size_t)(l0 + r) * D_ + d0 + threadIdx.x;
    z[idx] = x1[idx] * tile[threadIdx.x][r];
  }
}

// ---------------------------------------------------------------------------
extern "C" void kernel_launch(void* const* d_in, const int* in_sizes, int n_in,
                              void* d_out, int out_size, void* d_ws, size_t ws_size,
                              hipStream_t stream) {
  const float* x1 = (const float*)d_in[0];
  const float* x2 = (const float*)d_in[1];
  const float* v  = (const float*)d_in[2];
  const float* h  = (const float*)d_in[3];
  const float* cb = (const float*)d_in[4];
  float* out = (float*)d_out;

  // workspace carve-up (floats): KV | Y | Hfr | Hfi | Fc | Fs | Twc | Tws
  float* KV  = (float*)d_ws;
  float* Y   = KV + (size_t)B_ * D_ * L_;
  float* Hfr = Y + (size_t)B_ * D_ * L_;
  float* Hfi = Hfr + (size_t)G_ * N1_ * N1_;
  float* Fc  = Hfi + (size_t)G_ * N1_ * N1_;
  float* Fs  = Fc + N1_ * N1_;
  float* Twc = Fs + N1_ * N1_;
  float* Tws = Twc + N1_ * N1_;

  make_tables<<<64, 256, 0, stream>>>(Fc, Fs, Twc, Tws);
  kv_transpose<<<dim3(D_ / 32, L_ / 32, B_), dim3(32, 8), 0, stream>>>(x2, v, KV);
  // filter spectra: one block per group
  fftconv<<<G_, 256, 0, stream>>>(0, h, Fc, Fs, Twc, Tws, Hfr, Hfi, nullptr, nullptr);
  // main conv: one block per (b,d) sequence
  fftconv<<<B_ * D_, 256, 0, stream>>>(1, KV, Fc, Fs, Twc, Tws, Hfr, Hfi, cb, Y);
  gate_out<<<dim3(D_ / 32, L_ / 32, B_), dim3(32, 8), 0, stream>>>(x1, Y, out);
}